// StimulusSelfAttention_37409165148868
// MI455X (gfx1250) — compile-verified
//
#include <hip/hip_runtime.h>

// Per-row dot product: out[row] = sum_d x[row,d] * s[row,d]
// B=8, T=8192, D=768  ->  65,536 rows of 768 fp32 each. Pure HBM streaming.

typedef __attribute__((ext_vector_type(2))) float v2f;
typedef __attribute__((ext_vector_type(8))) float v8f;

#define D_DIM        768
#define VECS_PER_ROW (D_DIM / 4)        // 192 float4 per row
#define ITERS        (VECS_PER_ROW / 32) // 6 float4 loads per lane
#define WAVES_PER_BLOCK 8

__global__ __launch_bounds__(256) void rowdot_kernel(const float* __restrict__ x,
                                                     const float* __restrict__ s,
                                                     float* __restrict__ out,
                                                     int nrows) {
    const int lane = threadIdx.x & 31;
    const int wave = threadIdx.x >> 5;
    const int row  = blockIdx.x * WAVES_PER_BLOCK + wave;
    if (row >= nrows) return;  // wave-uniform: EXEC stays all-1s below

    const float4* __restrict__ xr =
        reinterpret_cast<const float4*>(x) + (size_t)row * VECS_PER_ROW;
    const float4* __restrict__ sr =
        reinterpret_cast<const float4*>(s) + (size_t)row * VECS_PER_ROW;

    // Coalesced streaming: lane l reads float4 #(32*i + l); 6 b128 loads per
    // input per lane, fp32 FMA accumulation in 4 independent chains.
    float a0 = 0.f, a1 = 0.f, a2 = 0.f, a3 = 0.f;
#pragma unroll
    for (int i = 0; i < ITERS; ++i) {
        float4 a = xr[i * 32 + lane];
        float4 b = sr[i * 32 + lane];
        a0 = fmaf(a.x, b.x, a0);
        a1 = fmaf(a.y, b.y, a1);
        a2 = fmaf(a.z, b.z, a2);
        a3 = fmaf(a.w, b.w, a3);
    }
    float p = (a0 + a1) + (a2 + a3);   // per-lane partial (24 elements)

    // Cross-lane reduction via exact-fp32 WMMA:
    //   A[16x4] holds the 32 partials (vgpr0 = p, vgpr1 = 0),
    //   B[4x16] = all ones  ->  D[m,n] = sum_k A[m,k]  (independent of n).
    // Summing the 8 D vgprs per lane and folding lane^16 covers every m
    // exactly once -> every lane holds the full 32-lane sum, exactly.
    v2f A;  A[0] = p;    A[1] = 0.0f;
    v2f Bv; Bv[0] = 1.0f; Bv[1] = 1.0f;
    v8f C = {};
    v8f Dm = __builtin_amdgcn_wmma_f32_16x16x4_f32(
        /*neg_a=*/false, A, /*neg_b=*/false, Bv,
        /*c_mod=*/(short)0, C, /*reuse_a=*/false, /*reuse_b=*/false);

    float t = ((Dm[0] + Dm[1]) + (Dm[2] + Dm[3])) +
              ((Dm[4] + Dm[5]) + (Dm[6] + Dm[7]));
    t += __shfl_xor(t, 16, 32);        // fold the two half-wave column sums

    if (lane == 0) out[row] = t;
}

extern "C" void kernel_launch(void* const* d_in, const int* in_sizes, int n_in,
                              void* d_out, int out_size, void* d_ws, size_t ws_size,
                              hipStream_t stream) {
    const float* x = (const float*)d_in[0];
    const float* s = (const float*)d_in[1];
    float* out     = (float*)d_out;

    const int nrows  = out_size;                        // B*T = 65536
    const int blocks = (nrows + WAVES_PER_BLOCK - 1) / WAVES_PER_BLOCK;
    rowdot_kernel<<<blocks, 32 * WAVES_PER_BLOCK, 0, stream>>>(x, s, out, nrows);
}